// VectorQuantizer_12627203850264
// MI455X (gfx1250) — compile-verified
//
#include <hip/hip_runtime.h>

typedef __attribute__((ext_vector_type(2))) float v2f;
typedef __attribute__((ext_vector_type(8))) float v8f;
typedef __attribute__((ext_vector_type(4))) int   v4i;

#define K_CODES     8192
#define DIM         64
#define NROWS       16384       // B*X*Y*Z
#define BATCH_STR   262144      // 64*4096 floats per batch in latents [B,D,X,Y,Z]
#define LOSS_OFF    262144      // loss scalar slot in d_out
#define LDS_STRIDE  68          // 64 + 4 pad -> conflict-free b64 reads
#define N_STAGES    128         // 8192 codebook rows / 64 per stage

#if defined(__has_builtin)
#  if __has_builtin(__builtin_amdgcn_global_load_async_to_lds_b128)
#    define HAVE_ASYNC_LDS 1
#  endif
#endif

#ifdef HAVE_ASYNC_LDS
typedef __attribute__((address_space(1))) v4i* gptr_v4i;   // global int4*
typedef __attribute__((address_space(3))) v4i* lptr_v4i;   // LDS int4*
__device__ __forceinline__ void async_copy16(const float* g, float* l) {
    // 16B per lane, memory -> LDS, tracked by ASYNCcnt
    __builtin_amdgcn_global_load_async_to_lds_b128((gptr_v4i)g, (lptr_v4i)l, 0, 0);
}
__device__ __forceinline__ void wait_async0() {
#  if __has_builtin(__builtin_amdgcn_s_wait_asynccnt)
    __builtin_amdgcn_s_wait_asynccnt(0);
#  else
    asm volatile("s_wait_asynccnt 0x0" ::: "memory");
#  endif
}
#endif

// ---------------------------------------------------------------------------
// Kernel 1: codebook row squared-norms, and zero the loss accumulator slot.
// ---------------------------------------------------------------------------
__global__ void vq_cnorm_kernel(const float* __restrict__ codebook,
                                float* __restrict__ cnorm,
                                float* __restrict__ loss_slot) {
    int i = blockIdx.x * blockDim.x + threadIdx.x;
    if (i == 0) *loss_slot = 0.0f;
    if (i < K_CODES) {
        const float* row = codebook + i * DIM;
        float s = 0.0f;
        #pragma unroll
        for (int d = 0; d < DIM; ++d) { float v = row[d]; s += v * v; }
        cnorm[i] = s;
    }
}

// ---------------------------------------------------------------------------
// Kernel 2: fused distance matmul (fp32 WMMA) + running per-row argmin.
// Each wave handles 16 rows; workgroup (8 waves) = 128 rows; grid = 128 blocks.
// Double-buffered LDS staging of 64 codebook rows per stage (async-to-LDS),
// one barrier per stage, 64 v_wmma_f32_16x16x4_f32 per stage per wave.
// ---------------------------------------------------------------------------
__global__ __launch_bounds__(256)
void vq_argmin_kernel(const float* __restrict__ latents,
                      const float* __restrict__ codebook,
                      const float* __restrict__ cnorm,
                      int* __restrict__ idxOut) {
    __shared__ float tile[2][64 * LDS_STRIDE];

    const int t    = threadIdx.x;
    const int wave = t >> 5;
    const int lane = t & 31;
    const int j    = lane & 15;   // N column within tile / M row for A
    const int h    = lane >> 4;   // half-wave: K pair select for A/B, M+8 for C/D

    const int rowbase = (blockIdx.x * 8 + wave) * 16;
    const int row     = rowbase + j;
    const int b       = row >> 12;
    const int s       = row & 4095;
    const float* latp = latents + b * BATCH_STR + s;

    // Preload A-tile (reused across all K): ISA 16x4 fp32 A layout.
    v2f areg[16];
    #pragma unroll
    for (int c = 0; c < 16; ++c) {
        const int d0 = 4 * c + 2 * h;
        areg[c].x = latp[(size_t)d0 * 4096];
        areg[c].y = latp[(size_t)(d0 + 1) * 4096];
    }

    float minv[8];
    int   mini[8];
    #pragma unroll
    for (int v = 0; v < 8; ++v) { minv[v] = 3.4e38f; mini[v] = 0; }

    const int ldrow = t >> 2;          // staging: 256 threads x 16 floats
    const int ldcol = (t & 3) * 16;
    float* lbase0 = &tile[0][ldrow * LDS_STRIDE + ldcol];
    float* lbase1 = &tile[1][ldrow * LDS_STRIDE + ldcol];

    // ---- prologue: stage 0 into tile[0] ----
    {
        const float* g = codebook + t * 16;
#ifdef HAVE_ASYNC_LDS
        #pragma unroll
        for (int q = 0; q < 4; ++q) async_copy16(g + q * 4, lbase0 + q * 4);
        wait_async0();
#else
        #pragma unroll
        for (int q = 0; q < 4; ++q)
            *reinterpret_cast<float4*>(lbase0 + q * 4) =
                *reinterpret_cast<const float4*>(g + q * 4);
#endif
    }
    __syncthreads();

    for (int stage = 0; stage < N_STAGES; ++stage) {
        const int  cur  = stage & 1;
        const bool more = (stage + 1) < N_STAGES;
        float* lnext = cur ? lbase0 : lbase1;

#ifdef HAVE_ASYNC_LDS
        // Issue next tile's async copies before compute; writes to the other
        // buffer conflict only with stage-1-old readers (one barrier behind).
        if (more) {
            const float* g = codebook + (size_t)(stage + 1) * (64 * DIM) + t * 16;
            #pragma unroll
            for (int q = 0; q < 4; ++q) async_copy16(g + q * 4, lnext + q * 4);
        }
#else
        float4 regs[4];
        if (more) {
            const float* g = codebook + (size_t)(stage + 1) * (64 * DIM) + t * 16;
            #pragma unroll
            for (int q = 0; q < 4; ++q)
                regs[q] = *reinterpret_cast<const float4*>(g + q * 4);
        }
#  if defined(__has_builtin)
#    if __has_builtin(__builtin_amdgcn_sched_barrier)
        __builtin_amdgcn_sched_barrier(0);   // keep prefetch loads issued early
#    endif
#  endif
#endif

        const float* tcur = &tile[cur][0];
        #pragma unroll
        for (int sub = 0; sub < 4; ++sub) {
            const int col = stage * 64 + sub * 16 + j;
            const float cn = cnorm[col];
            v8f acc = {0.f, 0.f, 0.f, 0.f, 0.f, 0.f, 0.f, 0.f};
            #pragma unroll
            for (int c = 0; c < 16; ++c) {
                v2f bvec = *reinterpret_cast<const v2f*>(
                    &tcur[(sub * 16 + j) * LDS_STRIDE + 4 * c + 2 * h]);
                acc = __builtin_amdgcn_wmma_f32_16x16x4_f32(
                    false, areg[c], false, bvec, (short)0, acc, false, false);
            }
            #pragma unroll
            for (int v = 0; v < 8; ++v) {
                const float sc = cn - 2.0f * acc[v];
                if (sc < minv[v]) { minv[v] = sc; mini[v] = col; }
            }
        }

#ifdef HAVE_ASYNC_LDS
        if (more) wait_async0();
#else
        if (more) {
            #pragma unroll
            for (int q = 0; q < 4; ++q)
                *reinterpret_cast<float4*>(lnext + q * 4) = regs[q];
        }
#endif
        __syncthreads();   // single barrier per stage (double-buffered)
    }

    // Cross-lane argmin over the 16 lanes of each half (tile columns).
    #pragma unroll
    for (int v = 0; v < 8; ++v) {
        float mv = minv[v];
        int   mi = mini[v];
        #pragma unroll
        for (int off = 1; off < 16; off <<= 1) {
            float ov = __shfl_xor(mv, off, 32);
            int   oi = __shfl_xor(mi, off, 32);
            if (ov < mv || (ov == mv && oi < mi)) { mv = ov; mi = oi; }
        }
        if (j == 0) idxOut[rowbase + 8 * h + v] = mi;
    }
}

// ---------------------------------------------------------------------------
// Kernel 3: gather codebook rows into [B,D,X,Y,Z] output + loss reduction.
// ---------------------------------------------------------------------------
__global__ __launch_bounds__(256)
void vq_gather_kernel(const float* __restrict__ latents,
                      const float* __restrict__ codebook,
                      const int* __restrict__ idx,
                      const float* __restrict__ wptr,
                      float* __restrict__ out,
                      float* __restrict__ loss_slot) {
    const int tid  = blockIdx.x * 256 + threadIdx.x;   // [0, 1048576)
    const int b    = tid >> 18;
    const int rest = tid & 262143;
    const int d    = rest >> 12;
    const int s    = rest & 4095;
    const int n    = b * 4096 + s;

    const float q  = codebook[(size_t)idx[n] * DIM + d];
    const float lv = latents[tid];
    out[tid] = q;

    const float diff = q - lv;
    float contrib = diff * diff;
    #pragma unroll
    for (int off = 16; off >= 1; off >>= 1)
        contrib += __shfl_xor(contrib, off, 32);

    if ((threadIdx.x & 31) == 0) {
        const float scale = (1.0f + *wptr) * (1.0f / 1048576.0f);
        atomicAdd(loss_slot, contrib * scale);
    }
}

// ---------------------------------------------------------------------------
extern "C" void kernel_launch(void* const* d_in, const int* in_sizes, int n_in,
                              void* d_out, int out_size, void* d_ws, size_t ws_size,
                              hipStream_t stream) {
    const float* latents  = (const float*)d_in[0];
    const float* vqweight = (const float*)d_in[1];
    const float* codebook = (const float*)d_in[2];

    float* out       = (float*)d_out;
    float* loss_slot = out + LOSS_OFF;
    float* cnorm     = (float*)d_ws;
    int*   idxArr    = (int*)((char*)d_ws + K_CODES * sizeof(float));

    vq_cnorm_kernel<<<K_CODES / 256, 256, 0, stream>>>(codebook, cnorm, loss_slot);
    vq_argmin_kernel<<<NROWS / 128, 256, 0, stream>>>(latents, codebook, cnorm, idxArr);
    vq_gather_kernel<<<(NROWS * DIM) / 256, 256, 0, stream>>>(latents, codebook, idxArr,
                                                              vqweight, out, loss_slot);
}